// CgsNodeFeat_79517024518205
// MI455X (gfx1250) — compile-verified
//
#include <hip/hip_runtime.h>
#include <hip/hip_bf16.h>
#include <math.h>

// ---------------------------------------------------------------------------
// MoNet-style Gaussian graph conv, restructured as:
//   proj[b,j,m,d] = sum_c feats[b,j,c] * conv_w[m,c,d]      (dense fp32 GEMM, WMMA)
//   out[b,n,m,d]  = relu( sum_k ew[b,n,k,m] * proj[b,idx,m,d] )
// Shapes: B=32 N=128 Knb=16 C=1024 KERNEL=8 D=128 (OUT=1024)
// ---------------------------------------------------------------------------

#define BATCH   32
#define NNODE   128
#define KNB     16
#define CDIM    1024
#define MKER    8
#define DDIM    128
#define BN      (BATCH * NNODE)      // 4096
#define TWO_PI_F 6.283185307179586f
#define EPS_F    1e-14f

typedef __attribute__((ext_vector_type(2))) float v2f;
typedef __attribute__((ext_vector_type(8))) float v8f;

// ---------------------------------------------------------------------------
// Kernel 1: proj = feats @ conv_w[m], fp32 WMMA 16x16x4 with 4-way row
// register blocking: each wave owns 4 row tiles (64 rows) x one 16-col tile,
// so one ds_load_2addr B-fragment feeds 4 WMMAs.
// grid = (DDIM/16, MKER, BN/512); block = 256 (8 wave32).
// LDS: B strip [1024 c][16 d], row stride 16 floats -> rows kk..kk+3 map to
// disjoint 16-bank groups, conflict-free fragment reads.
// ---------------------------------------------------------------------------
__global__ __launch_bounds__(256)
void proj_gemm_wmma(const float* __restrict__ feats,    // [BN, CDIM]
                    const float* __restrict__ conv_w,   // [MKER, CDIM, DDIM]
                    float* __restrict__ proj)           // [BN, MKER, DDIM]
{
    __shared__ float Bs[CDIM * 16];                     // 64 KB

    const int m    = blockIdx.y;
    const int col0 = blockIdx.x * 16;
    const int tid  = threadIdx.x;
    const int wave = tid >> 5;
    const int lane = tid & 31;
    const int row0 = blockIdx.z * 512 + wave * 64;      // 4 row tiles per wave

    // Stage B strip: Bs[c*16 + d] = conv_w[m][c][col0+d]  (coalesced 64B rows)
    const float* bsrc = conv_w + (size_t)m * CDIM * DDIM + col0;
    for (int i = tid; i < CDIM * 16; i += 256) {
        int c = i >> 4, d = i & 15;
        Bs[i] = bsrc[c * DDIM + d];
    }
    __syncthreads();

    // fp32 16x4 A-fragment layout: lanes 0-15 hold rows M=0..15;
    // VGPR0 = K=(lane>=16 ? 2:0), VGPR1 = K+1  -> one float2 load per lane.
    const int rlane = lane & 15;
    const int khalf = (lane >> 4) * 2;

    const float* arow[4];
#pragma unroll
    for (int t = 0; t < 4; ++t)
        arow[t] = feats + (size_t)(row0 + t * 16 + rlane) * CDIM + khalf;

    v8f acc[4] = {};

    for (int kk0 = 0; kk0 < CDIM; kk0 += 64) {
        if (kk0 + 64 < CDIM) {
#pragma unroll
            for (int t = 0; t < 4; ++t)                 // global_prefetch_b8
                __builtin_prefetch(arow[t] + kk0 + 64, 0, 0);
        }
#pragma unroll
        for (int k4 = 0; k4 < 64; k4 += 4) {
            const int kk = kk0 + k4;
            v2f b;
            b.x = Bs[(kk + khalf)     * 16 + rlane];
            b.y = Bs[(kk + khalf + 1) * 16 + rlane];
            v2f a0 = *(const v2f*)(arow[0] + kk);
            v2f a1 = *(const v2f*)(arow[1] + kk);
            v2f a2 = *(const v2f*)(arow[2] + kk);
            v2f a3 = *(const v2f*)(arow[3] + kk);
            acc[0] = __builtin_amdgcn_wmma_f32_16x16x4_f32(
                false, a0, false, b, (short)0, acc[0], false, false);
            acc[1] = __builtin_amdgcn_wmma_f32_16x16x4_f32(
                false, a1, false, b, (short)0, acc[1], false, false);
            acc[2] = __builtin_amdgcn_wmma_f32_16x16x4_f32(
                false, a2, false, b, (short)0, acc[2], false, false);
            acc[3] = __builtin_amdgcn_wmma_f32_16x16x4_f32(
                false, a3, false, b, (short)0, acc[3], false, false);
        }
    }

    // C/D layout: VGPR r -> M = r + (lane>=16 ? 8:0), N = lane&15
    const int rhalf = (lane >> 4) * 8;
#pragma unroll
    for (int t = 0; t < 4; ++t) {
#pragma unroll
        for (int r = 0; r < 8; ++r) {
            int row = row0 + t * 16 + r + rhalf;
            proj[(size_t)row * (MKER * DDIM) + m * DDIM + col0 + rlane] = acc[t][r];
        }
    }
}

// ---------------------------------------------------------------------------
// Kernel 2: Gaussian edge weights + neighbor aggregation + ReLU.
// grid = BN blocks; block = 128 threads (tid == output d).
// ---------------------------------------------------------------------------
__global__ __launch_bounds__(128)
void edge_aggregate(const float* __restrict__ centre,     // [B,N,2]
                    const int*   __restrict__ nidx,       // [B,N,KNB]
                    const float* __restrict__ gweights,   // [B,N,KNB]
                    const float* __restrict__ mean_rho,   // [8]
                    const float* __restrict__ mean_theta, // [8]
                    const float* __restrict__ prec_rho,   // [8]
                    const float* __restrict__ prec_theta, // [8]
                    const float* __restrict__ proj,       // [BN, MKER, DDIM]
                    float* __restrict__ out)              // [BN, MKER*DDIM]
{
    __shared__ int   s_idx[KNB];
    __shared__ float s_w [KNB][MKER];
    __shared__ float s_ew[KNB][MKER];

    const int bn  = blockIdx.x;          // b*N + n
    const int b   = bn >> 7;             // N = 128
    const int tid = threadIdx.x;

    if (tid < KNB) s_idx[tid] = nidx[bn * KNB + tid];
    __syncthreads();

    // One (k, m) pair per thread: 16*8 == 128.
    {
        const int k = tid >> 3, m = tid & 7;
        const int j = s_idx[k];
        const float cx = centre[bn * 2 + 0] - centre[(b * NNODE + j) * 2 + 0];
        const float cy = centre[bn * 2 + 1] - centre[(b * NNODE + j) * 2 + 1];
        const float rho   = sqrtf(cx * cx + cy * cy);
        const float theta = atan2f(cx, cy);            // note arg order (x, y)
        const float dr = rho - mean_rho[m];
        const float pr = prec_rho[m];
        const float wr = expf(-0.5f * dr * dr / (EPS_F + pr * pr));
        const float a1 = fabsf(theta - mean_theta[m]);
        const float a2 = fabsf(TWO_PI_F - a1);
        const float am = fminf(a1, a2);
        const float pt = prec_theta[m];
        const float wt = expf(-0.5f * am * am / (EPS_F + pt * pt));
        float w = wr * wt;
        if (w != w) w = 0.0f;                          // NaN -> 0
        s_w[k][m] = w;
    }
    __syncthreads();

    // Normalize over kernel dim, apply graph weight.
    {
        const int k = tid >> 3, m = tid & 7;
        float s = 0.0f;
#pragma unroll
        for (int mm = 0; mm < MKER; ++mm) s += s_w[k][mm];
        s_ew[k][m] = gweights[bn * KNB + k] * (s_w[k][m] / s);
    }
    __syncthreads();

    // Aggregate: tid == d; 16 coalesced 512B reads of L2-resident proj per m.
    const int base = b * NNODE;
#pragma unroll
    for (int m = 0; m < MKER; ++m) {
        float acc = 0.0f;
#pragma unroll
        for (int k = 0; k < KNB; ++k) {
            acc = fmaf(s_ew[k][m],
                       proj[(size_t)(base + s_idx[k]) * (MKER * DDIM) + m * DDIM + tid],
                       acc);
        }
        out[(size_t)bn * (MKER * DDIM) + m * DDIM + tid] = fmaxf(acc, 0.0f);
    }
}

// ---------------------------------------------------------------------------
extern "C" void kernel_launch(void* const* d_in, const int* in_sizes, int n_in,
                              void* d_out, int out_size, void* d_ws, size_t ws_size,
                              hipStream_t stream) {
    const float* node_feats = (const float*)d_in[0];  // [32,128,1024]
    const float* node_ctr   = (const float*)d_in[1];  // [32,128,2]
    const int*   nbr_idx    = (const int*)  d_in[2];  // [32,128,16]
    const float* gw         = (const float*)d_in[3];  // [32,128,16]
    const float* mean_rho   = (const float*)d_in[4];  // [1,8]
    const float* mean_theta = (const float*)d_in[5];
    const float* prec_rho   = (const float*)d_in[6];
    const float* prec_theta = (const float*)d_in[7];
    const float* conv_w     = (const float*)d_in[8];  // [8,1024,128]
    float*       out        = (float*)d_out;          // [32,128,1024]

    float* proj = (float*)d_ws;                       // BN*1024 floats = 16 MB

    dim3 g1(DDIM / 16, MKER, BN / 512);               // (8, 8, 8)
    proj_gemm_wmma<<<g1, 256, 0, stream>>>(node_feats, conv_w, proj);

    edge_aggregate<<<BN, 128, 0, stream>>>(node_ctr, nbr_idx, gw,
                                           mean_rho, mean_theta,
                                           prec_rho, prec_theta,
                                           proj, out);
}